// HierarchicalEmbedding_52475910422728
// MI455X (gfx1250) — compile-verified
//
#include <hip/hip_runtime.h>

typedef __attribute__((ext_vector_type(2))) float v2f;
typedef __attribute__((ext_vector_type(8))) float v8f;

#define LM1 6
#define DIM 128
#define TOK_PER_BLOCK 128
// Eint stored pair-interleaved: element [p*PAIR_W + d] = {row 2p, row 2p+1} at dim d.
// PAIR_W=144 float2 -> pair-row stride 288 dwords == 32 (mod 64): half-waves hit
// disjoint 32-bank groups on the b64 reads.
#define PAIR_W 144

__global__ __launch_bounds__(256)
void hier_embed_wmma(const int* __restrict__ first_chars,
                     const int* __restrict__ digits,
                     const int* __restrict__ ndigits,
                     const int* __restrict__ mods,
                     const float* __restrict__ vals,
                     const float* __restrict__ E1,
                     const float* __restrict__ Eint,
                     const int* __restrict__ kappa_p,
                     float* __restrict__ out,
                     int N)
{
    __shared__ v2f   s_eint2[6 * PAIR_W];   // K padded 10 -> 12 rows = 6 pair-rows
    __shared__ int   s_fidx[TOK_PER_BLOCK];
    __shared__ int   s_mod [TOK_PER_BLOCK];
    __shared__ float s_val [TOK_PER_BLOCK];
    __shared__ int   s_nd  [TOK_PER_BLOCK];
    __shared__ int   s_dig [TOK_PER_BLOCK * LM1];

    const int tid       = threadIdx.x;
    const int blockTok0 = blockIdx.x * TOK_PER_BLOCK;

    // ---- cooperative staging --------------------------------------------
    {
        float* ef = (float*)s_eint2;
        for (int i = tid; i < 12 * DIM; i += 256) {
            int k = i >> 7;        // source row (K)
            int d = i & (DIM - 1); // dim
            ef[((k >> 1) * PAIR_W + d) * 2 + (k & 1)] =
                (k < 10) ? Eint[k * DIM + d] : 0.0f;
        }
    }
    if (tid < TOK_PER_BLOCK) {
        int gT = blockTok0 + tid;
        if (gT > N - 1) gT = N - 1;
        int fc = first_chars[gT];
        int md = mods[gT];
        int fidx = (md == 0) ? (fc + 5) : ((md == 1) ? (fc * 2 + 5) : (52 + fc));
        s_fidx[tid] = fidx;
        s_mod[tid]  = md;
        s_val[tid]  = vals[gT];
        s_nd[tid]   = ndigits[gT];
    }
    {
        int maxi = N * LM1 - 1;
        for (int i = tid; i < TOK_PER_BLOCK * LM1; i += 256) {
            int idx = blockTok0 * LM1 + i;
            s_dig[i] = digits[(idx <= maxi) ? idx : maxi];
        }
    }
    __syncthreads();

    // ---- wave-local setup ------------------------------------------------
    const int wave  = tid >> 5;
    const int lane  = tid & 31;
    const int half  = lane >> 4;     // 0: lanes 0-15, 1: lanes 16-31
    const int nidx  = lane & 15;     // N-index within 16-wide tile / A-row M
    const int ltokA = wave * 16 + nidx;

    // position-decay weights 1/(l+2)^kappa, with the l<ndigits mask folded in
    // (branchless: avoids saveexec chains around the A-matrix build)
    const int kap = kappa_p[0];
    const int nd  = s_nd[ltokA];
    float mw[LM1];
    int   dg[LM1];
#pragma unroll
    for (int l = 0; l < LM1; ++l) {
        float p = 1.0f;
        for (int i = 0; i < kap; ++i) p *= (float)(l + 2);
        mw[l] = (l < nd) ? (1.0f / p) : 0.0f;
        dg[l] = s_dig[ltokA * LM1 + l];
    }

    v8f acc[8];
    {
        v8f z = {};
#pragma unroll
        for (int t = 0; t < 8; ++t) acc[t] = z;
    }

    // ---- digit einsum as f32 WMMA: [16x12] x [12x128], K-steps of 4 ------
#pragma unroll
    for (int s = 0; s < 3; ++s) {
        const int k0 = 4 * s + 2 * half;     // A: VGPR0 holds K=k0, VGPR1 K=k0+1
        float a0 = 0.0f, a1 = 0.0f;
#pragma unroll
        for (int l = 0; l < LM1; ++l) {
            a0 += (dg[l] == k0    ) ? mw[l] : 0.0f;   // v_cmp + v_cndmask + v_add
            a1 += (dg[l] == k0 + 1) ? mw[l] : 0.0f;
        }
        v2f A;
        A.x = a0;
        A.y = a1;
        const v2f* __restrict__ brow2 = s_eint2 + (k0 >> 1) * PAIR_W + nidx;
#pragma unroll
        for (int t = 0; t < 8; ++t) {
            v2f B = brow2[16 * t];           // single aligned ds_load_b64
            acc[t] = __builtin_amdgcn_wmma_f32_16x16x4_f32(
                /*neg_a=*/false, A, /*neg_b=*/false, B,
                /*c_mod=*/(short)0, acc[t], /*reuse_a=*/false, /*reuse_b=*/false);
        }
    }

    // ---- epilogue: base gather + modality select + store -----------------
#pragma unroll
    for (int r = 0; r < 8; ++r) {
        const int   lt   = wave * 16 + r + 8 * half;   // C row M=r(+8) -> token
        const int   gT   = blockTok0 + lt;
        const int   fidx = s_fidx[lt];
        const int   md   = s_mod[lt];
        const float vv   = s_val[lt];
        const bool  inb  = (gT < N);
        const float* __restrict__ brow = E1 + (size_t)fidx * DIM;
        float* __restrict__       orow = out + (size_t)gT * DIM;
#pragma unroll
        for (int t = 0; t < 8; ++t) {
            const int d = 16 * t + nidx;
            const float base = brow[d];
            const float res  = (md == 2) ? (base * vv) : (base + acc[t][r]);
            if (inb) orow[d] = res;
        }
    }
}

extern "C" void kernel_launch(void* const* d_in, const int* in_sizes, int n_in,
                              void* d_out, int out_size, void* d_ws, size_t ws_size,
                              hipStream_t stream) {
    const int*   first_chars = (const int*)  d_in[0];
    const int*   digits      = (const int*)  d_in[1];
    const int*   ndigits     = (const int*)  d_in[2];
    const int*   mods        = (const int*)  d_in[3];
    const float* vals        = (const float*)d_in[4];
    const float* E1          = (const float*)d_in[5];
    const float* Eint        = (const float*)d_in[6];
    const int*   kappa       = (const int*)  d_in[7];
    float*       out         = (float*)d_out;

    const int N = in_sizes[0];
    const int blocks = (N + TOK_PER_BLOCK - 1) / TOK_PER_BLOCK;
    hipLaunchKernelGGL(hier_embed_wmma, dim3(blocks), dim3(256), 0, stream,
                       first_chars, digits, ndigits, mods, vals, E1, Eint, kappa,
                       out, N);
}